// SeqProbMVSNet_72859825209773
// MI455X (gfx1250) — compile-verified
//
#include <hip/hip_runtime.h>
#include <hip/hip_bf16.h>

constexpr int V = 3, C = 16, D = 32, H = 256, W = 320;
constexpr int HW = H * W;

typedef __attribute__((ext_vector_type(2))) float v2f;
typedef __attribute__((ext_vector_type(8))) float v8f;

// ---------------------------------------------------------------------------
// Kernel A: projection setup (single thread; trivial cost).
// Computes for src views v=1,2:  M = P_v @ inv(P_0), stores rot(3x3)+trans(3)
// into ws[(v-1)*12 .. +11].
// ---------------------------------------------------------------------------
__global__ void proj_setup_kernel(const float* __restrict__ pm,
                                  float* __restrict__ ws) {
  if (threadIdx.x != 0 || blockIdx.x != 0) return;
  float P[3][16];
  for (int v = 0; v < 3; ++v) {
    const float* E = pm + v * 32;       // proj[:, v, 0] (4x4)
    const float* K = pm + v * 32 + 16;  // proj[:, v, 1] (4x4)
    for (int i = 0; i < 16; ++i) P[v][i] = E[i];
    for (int r = 0; r < 3; ++r)
      for (int c = 0; c < 4; ++c) {
        float s = 0.f;
        for (int k = 0; k < 3; ++k) s += K[r * 4 + k] * E[k * 4 + c];
        P[v][r * 4 + c] = s;
      }
  }
  // Gauss-Jordan inverse of P[0] with partial pivoting
  float a[4][8];
  for (int r = 0; r < 4; ++r)
    for (int c = 0; c < 4; ++c) {
      a[r][c] = P[0][r * 4 + c];
      a[r][c + 4] = (r == c) ? 1.f : 0.f;
    }
  for (int col = 0; col < 4; ++col) {
    int piv = col;
    float best = fabsf(a[col][col]);
    for (int r = col + 1; r < 4; ++r) {
      float m = fabsf(a[r][col]);
      if (m > best) { best = m; piv = r; }
    }
    if (piv != col)
      for (int c = 0; c < 8; ++c) {
        float t = a[col][c]; a[col][c] = a[piv][c]; a[piv][c] = t;
      }
    float inv = 1.f / a[col][col];
    for (int c = 0; c < 8; ++c) a[col][c] *= inv;
    for (int r = 0; r < 4; ++r)
      if (r != col) {
        float f = a[r][col];
        for (int c = 0; c < 8; ++c) a[r][c] -= f * a[col][c];
      }
  }
  float invRef[16];
  for (int r = 0; r < 4; ++r)
    for (int c = 0; c < 4; ++c) invRef[r * 4 + c] = a[r][c + 4];

  for (int v = 1; v < 3; ++v) {
    float M[16];
    for (int r = 0; r < 4; ++r)
      for (int c = 0; c < 4; ++c) {
        float s = 0.f;
        for (int k = 0; k < 4; ++k) s += P[v][r * 4 + k] * invRef[k * 4 + c];
        M[r * 4 + c] = s;
      }
    float* o = ws + (v - 1) * 12;
    for (int r = 0; r < 3; ++r) {
      o[r * 3 + 0] = M[r * 4 + 0];
      o[r * 3 + 1] = M[r * 4 + 1];
      o[r * 3 + 2] = M[r * 4 + 2];
    }
    o[9] = M[3]; o[10] = M[7]; o[11] = M[11];
  }
}

// ---------------------------------------------------------------------------
// Kernel B: warp + variance + WMMA-fused cost reduction.
// One wave32 handles 16 pixels for all 32 depths. Lane pair (l, l+16) owns
// pixel l%16 with channel-halves koff=0 / koff=2, matching the 16x16x4 F32
// WMMA A/B VGPR layouts exactly (no cross-lane data movement needed).
// ---------------------------------------------------------------------------
__global__ __launch_bounds__(256) void cost_kernel(
    const float* __restrict__ feat, const float* __restrict__ rt,
    const float* __restrict__ depth_values, const float* __restrict__ w_reg,
    float* __restrict__ cost_out) {
  const int lane  = threadIdx.x & 31;
  const int group = blockIdx.x * 8 + (threadIdx.x >> 5);  // 0..5119
  const int sub   = lane & 15;
  const int koff  = (lane >> 4) << 1;  // 0 for lanes 0-15, 2 for lanes 16-31
  const int pix   = group * 16 + sub;
  const float xf  = (float)(pix % W);
  const float yf  = (float)(pix / W);

  // rot @ [x,y,1] per src view (depth-independent, hoisted)
  float rx[2], ry[2], rz[2], tx[2], ty[2], tz[2];
#pragma unroll
  for (int s = 0; s < 2; ++s) {
    const float* m = rt + s * 12;
    rx[s] = m[0] * xf + m[1] * yf + m[2];
    ry[s] = m[3] * xf + m[4] * yf + m[5];
    rz[s] = m[6] * xf + m[7] * yf + m[8];
    tx[s] = m[9]; ty[s] = m[10]; tz[s] = m[11];
  }

  // Reference-view contribution (depth-independent) + A fragments (w_reg)
  float ref[8], refsq[8];
  v2f afrag[4];
#pragma unroll
  for (int j = 0; j < 4; ++j) {
    const int c0 = 4 * j + koff;
    ref[2 * j]     = feat[c0 * HW + pix];
    ref[2 * j + 1] = feat[(c0 + 1) * HW + pix];
    refsq[2 * j]     = ref[2 * j] * ref[2 * j];
    refsq[2 * j + 1] = ref[2 * j + 1] * ref[2 * j + 1];
    afrag[j].x = w_reg[c0];
    afrag[j].y = w_reg[c0 + 1];
  }

  for (int d = 0; d < D; ++d) {
    const float dv = depth_values[d * HW + pix];
    float sum[8], sq[8];
#pragma unroll
    for (int i = 0; i < 8; ++i) { sum[i] = ref[i]; sq[i] = refsq[i]; }

#pragma unroll
    for (int s = 0; s < 2; ++s) {
      float z = rz[s] * dv + tz[s];
      z = (fabsf(z) < 1e-6f) ? 1e-6f : z;
      const float invz = 1.0f / z;
      const float px = (rx[s] * dv + tx[s]) * invz;
      const float py = (ry[s] * dv + ty[s]) * invz;
      const float x0f = floorf(px), y0f = floorf(py);
      const float wx = px - x0f, wy = py - y0f;
      const int x0 = (int)x0f, y0 = (int)y0f;
      const float vx0 = (x0f >= 0.0f && x0f < (float)W) ? 1.0f : 0.0f;
      const float vx1 = (x0f + 1.0f >= 0.0f && x0f + 1.0f < (float)W) ? 1.0f : 0.0f;
      const float vy0 = (y0f >= 0.0f && y0f < (float)H) ? 1.0f : 0.0f;
      const float vy1 = (y0f + 1.0f >= 0.0f && y0f + 1.0f < (float)H) ? 1.0f : 0.0f;
      const int xc0 = min(max(x0, 0), W - 1);
      const int xc1 = min(max(x0 + 1, 0), W - 1);
      const int yc0 = min(max(y0, 0), H - 1);
      const int yc1 = min(max(y0 + 1, 0), H - 1);
      const float cw00 = (1.f - wx) * (1.f - wy) * vx0 * vy0;
      const float cw01 = wx * (1.f - wy) * vx1 * vy0;
      const float cw10 = (1.f - wx) * wy * vx0 * vy1;
      const float cw11 = wx * wy * vx1 * vy1;
      const int i00 = yc0 * W + xc0, i01 = yc0 * W + xc1;
      const int i10 = yc1 * W + xc0, i11 = yc1 * W + xc1;
      const float* fb = feat + (s + 1) * (C * HW);
#pragma unroll
      for (int i = 0; i < 8; ++i) {
        const int c = 4 * (i >> 1) + koff + (i & 1);
        const float* fc = fb + c * HW;
        const float g = cw00 * fc[i00] + cw01 * fc[i01] +
                        cw10 * fc[i10] + cw11 * fc[i11];
        sum[i] += g;
        sq[i] += g * g;
      }
    }

    // cost = w_reg . variance via 4 chained V_WMMA_F32_16X16X4_F32:
    // A = w chunks broadcast over rows, B = per-pixel variance chunks.
    v8f acc = {0.f, 0.f, 0.f, 0.f, 0.f, 0.f, 0.f, 0.f};
#pragma unroll
    for (int j = 0; j < 4; ++j) {
      const float s0 = sum[2 * j] * (1.0f / 3.0f);
      const float s1 = sum[2 * j + 1] * (1.0f / 3.0f);
      v2f bfrag;
      bfrag.x = sq[2 * j] * (1.0f / 3.0f) - s0 * s0;
      bfrag.y = sq[2 * j + 1] * (1.0f / 3.0f) - s1 * s1;
      acc = __builtin_amdgcn_wmma_f32_16x16x4_f32(
          false, afrag[j], false, bfrag, (short)0, acc, false, false);
    }
    // D[m][n] = cost[pixel n] for every m -> acc[0] holds this lane's cost.
    if (lane < 16) cost_out[d * HW + pix] = acc[0];
  }
}

// ---------------------------------------------------------------------------
// Kernel C: softmax over D, depth expectation, confidence. One thread/pixel.
// Reads costs from (and writes probs back to) the prob region of d_out.
// ---------------------------------------------------------------------------
__global__ __launch_bounds__(256) void softmax_kernel(
    const float* __restrict__ depth_values, float* __restrict__ out) {
  const int pix = blockIdx.x * 256 + threadIdx.x;
  if (pix >= HW) return;
  float* prob = out + 2 * HW;

  float cv[D];
  float m = -3.4e38f;
#pragma unroll
  for (int d = 0; d < D; ++d) {
    cv[d] = prob[d * HW + pix];
    m = fmaxf(m, cv[d]);
  }
  float s = 0.f;
#pragma unroll
  for (int d = 0; d < D; ++d) {
    cv[d] = __expf(cv[d] - m);
    s += cv[d];
  }
  const float inv = 1.0f / s;
  float depth = 0.f, didxf = 0.f;
#pragma unroll
  for (int d = 0; d < D; ++d) {
    const float p = cv[d] * inv;
    cv[d] = p;
    prob[d * HW + pix] = p;
    depth += p * depth_values[d * HW + pix];
    didxf += p * (float)d;
  }
  int didx = (int)didxf;
  didx = min(max(didx, 0), D - 1);
  float conf = 0.f;
#pragma unroll
  for (int d = 0; d < D; ++d)
    conf += (d == didx || d == didx + 1) ? cv[d] : 0.f;

  out[pix] = depth;
  out[HW + pix] = conf;
}

// ---------------------------------------------------------------------------
extern "C" void kernel_launch(void* const* d_in, const int* in_sizes, int n_in,
                              void* d_out, int out_size, void* d_ws,
                              size_t ws_size, hipStream_t stream) {
  const float* feat = (const float*)d_in[0];  // (V,B,C,H,W)
  const float* pm   = (const float*)d_in[1];  // (B,V,2,4,4)
  const float* dv   = (const float*)d_in[2];  // (B,D,H,W)
  const float* wr   = (const float*)d_in[3];  // (C,)
  float* out = (float*)d_out;                 // depth | conf | prob
  float* ws  = (float*)d_ws;

  proj_setup_kernel<<<1, 32, 0, stream>>>(pm, ws);
  // 5120 pixel-groups of 16, 8 waves (256 threads) per block -> 640 blocks
  cost_kernel<<<640, 256, 0, stream>>>(feat, ws, dv, wr, out + 2 * HW);
  softmax_kernel<<<(HW + 255) / 256, 256, 0, stream>>>(dv, out);
}